// SpatialAttentionBlock_50903952392626
// MI455X (gfx1250) — compile-verified
//
#include <hip/hip_runtime.h>
#include <math.h>

#define D_IN   64
#define D_EDGE 16
#define D_HID  64
#define HEADS  4
#define HD     256   // HEADS * D_HID
#define LN_EPS 1e-5f

typedef __attribute__((ext_vector_type(2))) float v2f;
typedef __attribute__((ext_vector_type(8))) float v8f;

__device__ __forceinline__ v8f wmma_f32_k4(v2f a, v2f b, v8f c) {
  // V_WMMA_F32_16X16X4_F32 : D = A(16x4) x B(4x16) + C(16x16), fp32 throughout
  return __builtin_amdgcn_wmma_f32_16x16x4_f32(false, a, false, b, (short)0, c,
                                               false, false);
}

// float atomic max via signed-max / unsigned-min ordering trick (amax init = -inf)
__device__ __forceinline__ void atomicMaxF(float* addr, float v) {
  if (v >= 0.0f)
    atomicMax((int*)addr, __float_as_int(v));
  else
    atomicMin((unsigned int*)addr, __float_as_uint(v));
}

// ---------------------------------------------------------------------------
// q = x @ W_node   [N,64] x [64,256] -> [N,256]
// one block = 16 rows, 4 waves; wave w owns cols [64w, 64w+64) = 4 16x16 tiles
// ---------------------------------------------------------------------------
__global__ void __launch_bounds__(128)
qproj_wmma(const float* __restrict__ X, const float* __restrict__ Wn,
           float* __restrict__ Q) {
  const int row0 = blockIdx.x * 16;
  const int wave = threadIdx.x >> 5;
  const int lane = threadIdx.x & 31;
  const int lr = lane & 15;        // row (A/D) or col (B/D) within tile
  const int lh = lane >> 4;        // lane half selects K pair / row half
  const int col0 = wave * 64;

  v8f acc0 = {}, acc1 = {}, acc2 = {}, acc3 = {};
  const float* xrow = X + (size_t)(row0 + lr) * D_IN;

  for (int k = 0; k < D_IN; k += 4) {
    v2f a;
    a.x = xrow[k + 2 * lh];
    a.y = xrow[k + 2 * lh + 1];
    const float* w0 = Wn + (size_t)(k + 2 * lh) * HD;
    const float* w1 = Wn + (size_t)(k + 2 * lh + 1) * HD;
    v2f b;
    b.x = w0[col0 + lr];      b.y = w1[col0 + lr];      acc0 = wmma_f32_k4(a, b, acc0);
    b.x = w0[col0 + 16 + lr]; b.y = w1[col0 + 16 + lr]; acc1 = wmma_f32_k4(a, b, acc1);
    b.x = w0[col0 + 32 + lr]; b.y = w1[col0 + 32 + lr]; acc2 = wmma_f32_k4(a, b, acc2);
    b.x = w0[col0 + 48 + lr]; b.y = w1[col0 + 48 + lr]; acc3 = wmma_f32_k4(a, b, acc3);
  }
#pragma unroll
  for (int i = 0; i < 8; ++i) {
    const int r = row0 + i + 8 * lh;           // C/D layout: VGPR i -> M = i + 8*laneHalf
    float* qr = Q + (size_t)r * HD + col0 + lr;
    qr[0]  = acc0[i];
    qr[16] = acc1[i];
    qr[32] = acc2[i];
    qr[48] = acc3[i];
  }
}

// ---------------------------------------------------------------------------
// init amax = -inf
// ---------------------------------------------------------------------------
__global__ void init_amax(float* __restrict__ amax, int n) {
  int i = blockIdx.x * 256 + threadIdx.x;
  if (i < n) amax[i] = -__builtin_huge_valf();
}

// ---------------------------------------------------------------------------
// Edge pass 1: one wave per edge. Lane owns 8 of 256 message dims.
// msg = q[src] + edge_attr @ W_edge ; alpha = dot(q[dst], msg)/8 per head.
// Store raw alpha, atomic-max into amax[dst].
// ---------------------------------------------------------------------------
__device__ __forceinline__ void edge_msg(const float* __restrict__ ea,
                                         const float* __restrict__ We,
                                         const float* __restrict__ q,
                                         int e, int s, int d0, float msg[8]) {
  const float4* eap = (const float4*)(ea + (size_t)e * D_EDGE);
  float4 t0 = eap[0], t1 = eap[1], t2 = eap[2], t3 = eap[3];
  float ef[16] = {t0.x, t0.y, t0.z, t0.w, t1.x, t1.y, t1.z, t1.w,
                  t2.x, t2.y, t2.z, t2.w, t3.x, t3.y, t3.z, t3.w};
#pragma unroll
  for (int j = 0; j < 8; ++j) {
    float a = 0.0f;
#pragma unroll
    for (int k = 0; k < 16; ++k) a = fmaf(ef[k], We[k * HD + d0 + j], a);
    msg[j] = a;
  }
  const float4* qs = (const float4*)(q + (size_t)s * HD + d0);
  float4 s0 = qs[0], s1 = qs[1];
  msg[0] += s0.x; msg[1] += s0.y; msg[2] += s0.z; msg[3] += s0.w;
  msg[4] += s1.x; msg[5] += s1.y; msg[6] += s1.z; msg[7] += s1.w;
}

__global__ void __launch_bounds__(256)
edge_alpha(const float* __restrict__ ea, const float* __restrict__ We,
           const float* __restrict__ q, const int* __restrict__ src,
           const int* __restrict__ dst, float* __restrict__ alpha_raw,
           float* __restrict__ amax, int E) {
  const int lane = threadIdx.x & 31;
  const int e = blockIdx.x * 8 + (threadIdx.x >> 5);
  if (e >= E) return;
  const int s = src[e], d = dst[e];
  const int d0 = lane * 8;

  float msg[8];
  edge_msg(ea, We, q, e, s, d0, msg);

  const float4* qdp = (const float4*)(q + (size_t)d * HD + d0);
  float4 u0 = qdp[0], u1 = qdp[1];
  float p = u0.x * msg[0] + u0.y * msg[1] + u0.z * msg[2] + u0.w * msg[3] +
            u1.x * msg[4] + u1.y * msg[5] + u1.z * msg[6] + u1.w * msg[7];
  // reduce across the 8 lanes of this head
  p += __shfl_xor(p, 1, 32);
  p += __shfl_xor(p, 2, 32);
  p += __shfl_xor(p, 4, 32);
  if ((lane & 7) == 0) {
    const int h = lane >> 3;
    const float al = p * 0.125f;  // 1/sqrt(64)
    alpha_raw[(size_t)e * HEADS + h] = al;
    atomicMaxF(&amax[(size_t)d * HEADS + h], al);
  }
}

// ---------------------------------------------------------------------------
// Softmax numerator + denominator: ex = exp(alpha - amax[dst]); denom += ex
// ---------------------------------------------------------------------------
__global__ void __launch_bounds__(256)
edge_softmax(float* __restrict__ ex_io, const float* __restrict__ amax,
             const int* __restrict__ dst, float* __restrict__ denom, int E) {
  const int i = blockIdx.x * 256 + threadIdx.x;
  if (i >= E * HEADS) return;
  const int e = i >> 2, h = i & 3;
  const int d = dst[e];
  const float v = __expf(ex_io[i] - amax[(size_t)d * HEADS + h]);
  ex_io[i] = v;
  unsafeAtomicAdd(&denom[(size_t)d * HEADS + h], v);
}

// ---------------------------------------------------------------------------
// Edge pass 2: recompute msg, scale by ex/denom[dst], scatter-add into accum
// ---------------------------------------------------------------------------
__global__ void __launch_bounds__(256)
edge_aggr(const float* __restrict__ ea, const float* __restrict__ We,
          const float* __restrict__ q, const int* __restrict__ src,
          const int* __restrict__ dst, const float* __restrict__ exb,
          const float* __restrict__ denom, float* __restrict__ accum, int E) {
  const int lane = threadIdx.x & 31;
  const int e = blockIdx.x * 8 + (threadIdx.x >> 5);
  if (e >= E) return;
  const int s = src[e], d = dst[e];
  const int d0 = lane * 8;
  const int h = lane >> 3;

  float msg[8];
  edge_msg(ea, We, q, e, s, d0, msg);

  const float w = exb[(size_t)e * HEADS + h] / denom[(size_t)d * HEADS + h];
  float* outp = accum + (size_t)d * HD + d0;
#pragma unroll
  for (int j = 0; j < 8; ++j) unsafeAtomicAdd(&outp[j], msg[j] * w);
}

// ---------------------------------------------------------------------------
// y = LN(accum @ W_out + b_out + x) ; WMMA K=256 then LDS-based LayerNorm
// one block = 16 rows; 4 waves, wave w owns output cols [16w, 16w+16)
// ---------------------------------------------------------------------------
__global__ void __launch_bounds__(128)
out_ln_wmma(const float* __restrict__ A, const float* __restrict__ Wo,
            const float* __restrict__ bo, const float* __restrict__ Xres,
            const float* __restrict__ gamma, const float* __restrict__ beta,
            float* __restrict__ Y) {
  __shared__ float lds[16 * 64];
  const int row0 = blockIdx.x * 16;
  const int wave = threadIdx.x >> 5;
  const int lane = threadIdx.x & 31;
  const int lr = lane & 15;
  const int lh = lane >> 4;
  const int col0 = wave * 16;

  v8f acc = {};
  const float* arow = A + (size_t)(row0 + lr) * HD;
  for (int k = 0; k < HD; k += 4) {
    v2f a, b;
    a.x = arow[k + 2 * lh];
    a.y = arow[k + 2 * lh + 1];
    b.x = Wo[(size_t)(k + 2 * lh) * D_HID + col0 + lr];
    b.y = Wo[(size_t)(k + 2 * lh + 1) * D_HID + col0 + lr];
    acc = wmma_f32_k4(a, b, acc);
  }
#pragma unroll
  for (int i = 0; i < 8; ++i)
    lds[(i + 8 * lh) * 64 + col0 + lr] = acc[i];
  __syncthreads();

  // bias + residual
  for (int idx = threadIdx.x; idx < 16 * 64; idx += 128) {
    const int r = idx >> 6, c = idx & 63;
    lds[idx] += bo[c] + Xres[(size_t)(row0 + r) * 64 + c];
  }
  __syncthreads();

  // LayerNorm: one thread per row (16 rows, 64 cols each)
  if (threadIdx.x < 16) {
    const int r = threadIdx.x;
    float mu = 0.0f;
    for (int c = 0; c < 64; ++c) mu += lds[r * 64 + c];
    mu *= (1.0f / 64.0f);
    float var = 0.0f;
    for (int c = 0; c < 64; ++c) {
      const float dd = lds[r * 64 + c] - mu;
      var += dd * dd;
    }
    var *= (1.0f / 64.0f);
    const float rs = rsqrtf(var + LN_EPS);
    for (int c = 0; c < 64; ++c)
      Y[(size_t)(row0 + r) * 64 + c] =
          (lds[r * 64 + c] - mu) * rs * gamma[c] + beta[c];
  }
}

// ---------------------------------------------------------------------------
extern "C" void kernel_launch(void* const* d_in, const int* in_sizes, int n_in,
                              void* d_out, int out_size, void* d_ws, size_t ws_size,
                              hipStream_t stream) {
  const float* x     = (const float*)d_in[0];  // [N,64]
  const float* ea    = (const float*)d_in[1];  // [E,16]
  const float* Wn    = (const float*)d_in[2];  // [64,256]
  const float* We    = (const float*)d_in[3];  // [16,256]
  const float* Wo    = (const float*)d_in[4];  // [256,64]
  const float* bo    = (const float*)d_in[5];  // [64]
  const float* gamma = (const float*)d_in[6];  // [64]
  const float* beta  = (const float*)d_in[7];  // [64]
  const int*   ei    = (const int*)d_in[8];    // [2,E]

  const int N = in_sizes[0] / D_IN;
  const int E = in_sizes[8] / 2;
  const int* src = ei;
  const int* dst = ei + E;

  // workspace layout (floats): q[N*256] | accum[N*256] | amax[N*4] | denom[N*4] | ex[E*4]
  float* q     = (float*)d_ws;
  float* accum = q + (size_t)N * HD;
  float* amax  = accum + (size_t)N * HD;
  float* denom = amax + (size_t)N * HEADS;
  float* exb   = denom + (size_t)N * HEADS;

  hipMemsetAsync(accum, 0, (size_t)N * HD * sizeof(float), stream);
  hipMemsetAsync(denom, 0, (size_t)N * HEADS * sizeof(float), stream);
  init_amax<<<(N * HEADS + 255) / 256, 256, 0, stream>>>(amax, N * HEADS);

  qproj_wmma<<<N / 16, 128, 0, stream>>>(x, Wn, q);
  edge_alpha<<<(E + 7) / 8, 256, 0, stream>>>(ea, We, q, src, dst, exb, amax, E);
  edge_softmax<<<(E * HEADS + 255) / 256, 256, 0, stream>>>(exb, amax, dst, denom, E);
  edge_aggr<<<(E + 7) / 8, 256, 0, stream>>>(ea, We, q, src, dst, exb, denom, accum, E);
  out_ln_wmma<<<N / 16, 128, 0, stream>>>(accum, Wo, bo, x, gamma, beta, (float*)d_out);
}